// SplashEncoding_88141318849078
// MI455X (gfx1250) — compile-verified
//
#include <hip/hip_runtime.h>
#include <hip/hip_bf16.h>
#include <float.h>

// -------- problem constants (from reference) --------
#define MQ    32768
#define NG    10000
#define NPAD  10240          // padded candidate count (40 chunks of 256)
#define FF    32
#define KN    16
#define CHUNK 256
#define NCHUNK (NPAD / CHUNK) // 40
#define EPSW  1e-8f

typedef __attribute__((ext_vector_type(2))) float    v2f;
typedef __attribute__((ext_vector_type(8))) float    v8f;
typedef __attribute__((ext_vector_type(4))) unsigned u32x4;
typedef __attribute__((ext_vector_type(4))) int      i32x4;
typedef __attribute__((ext_vector_type(8))) int      i32x8;

// ---------------------------------------------------------------------------
// Tensor Data Mover: 1-D tile (nelem 4-byte words) global -> LDS.
// D# group0: count=1, lds_addr, global_addr[56:0], type=2.
// D# group1: data_size=4B, tensor_dim0=tile_dim0=nelem, stride=nelem.
// ---------------------------------------------------------------------------
#if __has_builtin(__builtin_amdgcn_tensor_load_to_lds)
#define HAVE_TDM 1
__device__ __forceinline__ void tdm_load_1d(unsigned lds_addr, const void* gptr,
                                            unsigned nelem /* 4B units, <=65535 */) {
    unsigned long long ga = (unsigned long long)(uintptr_t)gptr;
    u32x4 g0;
    g0[0] = 1u;                                              // count=1 (valid), user mode
    g0[1] = lds_addr;                                        // lds_addr[31:0]
    g0[2] = (unsigned)ga;                                    // global_addr[31:0]
    g0[3] = (unsigned)((ga >> 32) & 0x01ffffffu) | (2u << 30); // addr[56:32] | type=2
    i32x8 g1;
    g1[0] = 0x00020000;                                      // data_size=2 (4 bytes), mask=0
    g1[1] = (int)((nelem & 0xffffu) << 16);                  // tensor_dim0[15:0] @ bits63:48
    g1[2] = (int)(((nelem >> 16) & 0xffffu) | (1u << 16));   // tensor_dim0[31:16] | tensor_dim1=1
    g1[3] = (int)((nelem & 0xffffu) << 16);                  // tile_dim0 @ bits127:112
    g1[4] = 0;                                               // tile_dim1=0, tile_dim2=0
    g1[5] = (int)nelem;                                      // tensor_dim0_stride lo
    g1[6] = 0;
    g1[7] = 0;
    i32x4 z4 = {};
#if defined(__clang_major__) && __clang_major__ >= 23
    i32x8 z8 = {};
    __builtin_amdgcn_tensor_load_to_lds(g0, g1, z4, z4, z8, 0);
#else
    __builtin_amdgcn_tensor_load_to_lds(g0, g1, z4, z4, 0);
#endif
}
#endif

// ---------------------------------------------------------------------------
// Kernel 0: pack candidates for the WMMA score GEMM.
//   pm[n] = [-2*mx, -2*my, -2*mz, |m|^2]   (pad rows get score +FLT_MAX)
// score(q,n) = dot([cx,cy,cz,1], pm[n]) = d2(q,n) - |c|^2  (row-constant shift
// leaves per-row top-k unchanged).
// ---------------------------------------------------------------------------
__global__ __launch_bounds__(256)
void splash_pack(const float* __restrict__ means, float4* __restrict__ pm) {
    int i = blockIdx.x * blockDim.x + threadIdx.x;
    if (i >= NPAD) return;
    if (i < NG) {
        float mx = means[i * 3 + 0];
        float my = means[i * 3 + 1];
        float mz = means[i * 3 + 2];
        pm[i] = make_float4(-2.f * mx, -2.f * my, -2.f * mz,
                            mx * mx + my * my + mz * mz);
    } else {
        pm[i] = make_float4(0.f, 0.f, 0.f, FLT_MAX);  // never selected
    }
}

// ---------------------------------------------------------------------------
// Kernel 1: fused KNN (WMMA score tiles + per-lane replace-max top-16) +
// gaussian splash aggregation. Block = 256 threads = 8 waves; each wave owns
// 32 query rows (two 16-row WMMA tiles), each lane owns one query row.
// Candidate chunks are DMA'd to LDS by the TDM, double-buffered.
// ---------------------------------------------------------------------------
__global__ __launch_bounds__(256)
void splash_knn_agg(const float* __restrict__ coords,
                    const float* __restrict__ means,
                    const float* __restrict__ log_covs,
                    const float* __restrict__ feats,
                    const float4* __restrict__ pm,
                    float* __restrict__ out) {
    __shared__ __align__(16) float4 s_pm[2][CHUNK];          // double-buffered chunks (8 KB)
    __shared__ __align__(16) float  s_tile[8][2][16][20];    // per-wave score tiles (20 KB)

    const int tid  = threadIdx.x;
    const int wave = tid >> 5;
    const int lane = tid & 31;
    const int half = lane >> 4;   // 0 = lanes 0-15, 1 = lanes 16-31
    const int lrow = lane & 15;

    const int wavebase = blockIdx.x * 256 + wave * 32;

    // ---- A fragments: two 16x4 tiles of [cx,cy,cz,1] ----
    // V_WMMA_F32_16X16X4_F32 A layout: lanes 0-15 hold K=0,1; lanes 16-31 hold K=2,3.
    v2f aA, aB;
    {
        int qa = wavebase + lrow;
        int qb = qa + 16;
        if (half == 0) {
            aA[0] = coords[qa * 3 + 0]; aA[1] = coords[qa * 3 + 1];
            aB[0] = coords[qb * 3 + 0]; aB[1] = coords[qb * 3 + 1];
        } else {
            aA[0] = coords[qa * 3 + 2]; aA[1] = 1.0f;
            aB[0] = coords[qb * 3 + 2]; aB[1] = 1.0f;
        }
    }

    // ---- per-lane top-K: unsorted list + tracked worst (branchless update) ----
    float best_d[KN];
    int   best_i[KN];
#pragma unroll
    for (int j = 0; j < KN; ++j) { best_d[j] = FLT_MAX; best_i[j] = 0; }
    float worst = FLT_MAX;

    auto consider = [&](float d, int idx) {
        if (d < worst) {                      // early-out vs current k-th best
            bool done = false;
#pragma unroll
            for (int j = 0; j < KN; ++j) {    // replace first slot holding worst
                bool repl = (!done) && (best_d[j] == worst);
                if (repl) { best_d[j] = d; best_i[j] = idx; }
                done = done || repl;
            }
            float nw = best_d[0];             // recompute worst (v_max tree)
#pragma unroll
            for (int j = 1; j < KN; ++j) nw = fmaxf(nw, best_d[j]);
            worst = nw;
        }
    };

#ifdef HAVE_TDM
    if (wave == 0)                            // prologue: DMA chunk 0
        tdm_load_1d((unsigned)(uintptr_t)&s_pm[0][0], pm, CHUNK * 4);
#endif

    for (int ch = 0; ch < NCHUNK; ++ch) {
        const int buf = ch & 1;
#ifdef HAVE_TDM
        if (wave == 0)
            __builtin_amdgcn_s_wait_tensorcnt(0);   // chunk ch resident in LDS
        __syncthreads();                            // visibility + all prev reads done
        if (wave == 0 && ch + 1 < NCHUNK)           // overlap: DMA next chunk
            tdm_load_1d((unsigned)(uintptr_t)&s_pm[buf ^ 1][0],
                        pm + (ch + 1) * CHUNK, CHUNK * 4);
#else
        __syncthreads();
        s_pm[buf][tid & (CHUNK - 1)] = pm[ch * CHUNK + (tid & (CHUNK - 1))];
        __syncthreads();
#endif

#pragma unroll 1
        for (int sub = 0; sub < CHUNK / 16; ++sub) {
            // B fragment: 4x16 tile. Lanes 0-15 hold K=0,1; lanes 16-31 K=2,3.
            float4 p = s_pm[buf][sub * 16 + lrow];
            v2f b;
            if (half == 0) { b[0] = p.x; b[1] = p.y; }
            else           { b[0] = p.z; b[1] = p.w; }

            v8f cA = {}; v8f cB = {};
            // args: (neg_a, A, neg_b, B, c_mod, C, reuse_a, reuse_b)
            cA = __builtin_amdgcn_wmma_f32_16x16x4_f32(
                     false, aA, false, b, (short)0, cA, false, false);
            cB = __builtin_amdgcn_wmma_f32_16x16x4_f32(
                     false, aB, false, b, (short)0, cB, false, false);

            // C layout: VGPR r / lanes 0-15 -> row r; lanes 16-31 -> row r+8;
            // col = lane&15.  Transpose through LDS (rows 16B-aligned, stride 20).
#pragma unroll
            for (int r = 0; r < 8; ++r) {
                s_tile[wave][0][half * 8 + r][lrow] = cA[r];
                s_tile[wave][1][half * 8 + r][lrow] = cB[r];
            }
            asm volatile("s_wait_dscnt 0" ::: "memory");

            // lane -> (tile=half, row=lrow) i.e. query wavebase+lane
            const int nbase = ch * CHUNK + sub * 16;
            const float4* row4 = (const float4*)&s_tile[wave][half][lrow][0];
#pragma unroll
            for (int cc = 0; cc < 4; ++cc) {
                float4 v = row4[cc];
                consider(v.x, nbase + 4 * cc + 0);
                consider(v.y, nbase + 4 * cc + 1);
                consider(v.z, nbase + 4 * cc + 2);
                consider(v.w, nbase + 4 * cc + 3);
            }
            asm volatile("s_wait_dscnt 0" ::: "memory");
        }
    }

    // ---- park top-K indices in this wave's own tile region (no cross-wave
    // traffic, same-wave DS ordering => no barrier needed) ----
    int* s_idx = (int*)&s_tile[wave][0][0][0];   // 640 slots >= 32*16
#pragma unroll
    for (int j = 0; j < KN; ++j) s_idx[lane * KN + j] = best_i[j];
    asm volatile("s_wait_dscnt 0" ::: "memory");

    // ---- gaussian splash aggregation for this lane's query ----
    const int q = wavebase + lane;
    const float cx = coords[q * 3 + 0];
    const float cy = coords[q * 3 + 1];
    const float cz = coords[q * 3 + 2];

    float acc[FF];
#pragma unroll
    for (int f = 0; f < FF; ++f) acc[f] = 0.f;
    float wsum = 0.f;

#pragma unroll 1
    for (int k2 = 0; k2 < KN; ++k2) {
        int n = s_idx[lane * KN + k2];
        float dx = cx - means[n * 3 + 0];
        float dy = cy - means[n * 3 + 1];
        float dz = cz - means[n * 3 + 2];
        float vx = expf(log_covs[n * 3 + 0]);
        float vy = expf(log_covs[n * 3 + 1]);
        float vz = expf(log_covs[n * 3 + 2]);
        float logw = -0.5f * (dx * dx / vx + dy * dy / vy + dz * dz / vz);
        float w = expf(logw);
        wsum += w;

        const float4* f4 = (const float4*)(feats + (size_t)n * FF);
#pragma unroll
        for (int j = 0; j < 8; ++j) {
            float4 fv = f4[j];
            acc[4 * j + 0] += w * fv.x;
            acc[4 * j + 1] += w * fv.y;
            acc[4 * j + 2] += w * fv.z;
            acc[4 * j + 3] += w * fv.w;
        }
    }

    const float inv = 1.0f / (wsum + EPSW);
    float4* o4 = (float4*)(out + (size_t)q * FF);
#pragma unroll
    for (int j = 0; j < 8; ++j) {
        o4[j] = make_float4(acc[4 * j + 0] * inv, acc[4 * j + 1] * inv,
                            acc[4 * j + 2] * inv, acc[4 * j + 3] * inv);
    }
}

// ---------------------------------------------------------------------------
extern "C" void kernel_launch(void* const* d_in, const int* in_sizes, int n_in,
                              void* d_out, int out_size, void* d_ws, size_t ws_size,
                              hipStream_t stream) {
    const float* coords   = (const float*)d_in[0];   // [M,3]
    const float* means    = (const float*)d_in[1];   // [N,3]
    const float* log_covs = (const float*)d_in[2];   // [N,3]
    const float* feats    = (const float*)d_in[3];   // [N,F]
    // d_in[4] = k (fixed at 16, compile-time here)
    float*  out = (float*)d_out;                     // [M,F]
    float4* pm  = (float4*)d_ws;                     // NPAD packed candidates (160 KB)

    splash_pack<<<NPAD / 256, 256, 0, stream>>>(means, pm);
    splash_knn_agg<<<MQ / 256, 256, 0, stream>>>(coords, means, log_covs, feats, pm, out);
}